// TopDownLSTM_37615323579253
// MI455X (gfx1250) — compile-verified
//
#include <hip/hip_runtime.h>

// ---------------------------------------------------------------------------
// TopDownLSTM for MI455X (gfx1250, wave32, WMMA).
//
//  * All matmuls use V_WMMA_F32_16X16X4_F32 (native fp32 matrix op) for
//    reference-grade precision over 256 recurrent timesteps.
//  * Weights (~126 MB fp32) are L2-resident (192 MB L2); the per-step GEMMs
//    stream weights from L2, so no precision downconversion is warranted.
//  * The t-loop is inherently sequential: 3 graph-captured kernels per step.
//  * Latency optimization: K reduction is split 4-ways across waves (16 waves
//    per block = 4 gates x 4 K-slices) and reduced through the same LDS tile
//    used for the f/i/o/g gate recombine. Compile-time K -> immediate-offset
//    load clauses instead of per-iteration 64-bit address arithmetic.
//  * Ping-pong h/c state buffers in d_ws give race-free in-place semantics.
// ---------------------------------------------------------------------------

#define T_STEPS 256
#define B_SIZE  64
#define I_SIZE  512
#define H_SIZE  1024
#define ZCOLS   4096            // 4*H
#define BH      (B_SIZE * H_SIZE)

typedef __attribute__((ext_vector_type(2))) float v2f;
typedef __attribute__((ext_vector_type(8))) float v8f;

__device__ __forceinline__ float sigmoidf_(float x) {
    return 1.0f / (1.0f + __expf(-x));
}

// Accumulate a 64x16 z-tile over a compile-time K-slice.
//   A : pointer to A[0][k0]      (row stride KROW, row-major [64 x KROW])
//   W : pointer to W[k0][n=0]    (row stride ZCOLS)
// Fragment layouts per ISA 7.12.2 (32-bit, wave32):
//   A 16x4 : lane = M row (lc), {VGPR,half} = K  -> contiguous float2/lane
//   B 4x16 : lane = N col (lc), {VGPR,half} = K  (mirror of A)
template <int KQ, int KROW>
__device__ __forceinline__ void gemm_term(const float* __restrict__ A,
                                          const float* __restrict__ W,
                                          v8f acc[4], int lc, int kh, int n0)
{
    const float* __restrict__ Wp = W + (size_t)(kh << 1) * ZCOLS + n0 + lc;
    const float* __restrict__ Ap = A + (kh << 1) + lc * KROW;
    #pragma unroll 4
    for (int k = 0; k < KQ; k += 4) {
        v2f bfrag;
        bfrag.x = Wp[(size_t)k * ZCOLS];
        bfrag.y = Wp[(size_t)k * ZCOLS + ZCOLS];
        #pragma unroll
        for (int mb = 0; mb < 4; ++mb) {
            v2f afrag = *(const v2f*)(Ap + (size_t)(mb * 16) * KROW + k);
            acc[mb] = __builtin_amdgcn_wmma_f32_16x16x4_f32(
                /*neg_a=*/false, afrag, /*neg_b=*/false, bfrag,
                /*c_mod=*/(short)0, acc[mb],
                /*reuse_a=*/false, /*reuse_b=*/false);
        }
    }
}

// One LSTM layer for one timestep.
//   z[64 x 4096] = sum_s A_s[64 x K_s] @ W_s[K_s x 4096] + bias
// Gate column layout (reference split order): f | i | o | g, 1024 cols each.
//
// Grid:  64 blocks   -> gate-column slab [16*bx, 16*bx+16)
// Block: 512 threads -> 16 waves; wave = (kslice, gate). Each wave runs 1/4 of
//        the K reduction for its gate's 64x16 tile; slices are summed in the
//        LDS gate-recombine phase.
// KQ_s = K_s / 4 per source (KQ2 == 0 -> source absent).
template <int KQ0, int KQ1, int KQ2, bool WRITE_OUT>
__global__ __launch_bounds__(512)
void lstm_layer_kernel(const float* __restrict__ A0, const float* __restrict__ W0,
                       const float* __restrict__ A1, const float* __restrict__ W1,
                       const float* __restrict__ A2, const float* __restrict__ W2,
                       const float* __restrict__ bias,
                       const float* __restrict__ h_prev, const float* __restrict__ c_prev,
                       float* __restrict__ h_out, float* __restrict__ c_out,
                       float* __restrict__ out_t,                    // output[t] (layer 2)
                       const int* __restrict__ length, int t)
{
    __shared__ float zlds[4][4][64][17];   // [kslice][gate][row][16 cols + pad]

    const int tid  = threadIdx.x;
    const int wave = tid >> 5;
    const int gate = wave & 3;             // f,i,o,g
    const int kq   = wave >> 2;            // K-slice 0..3
    const int lane = tid & 31;
    const int lc   = lane & 15;            // M-row / N-col lane index
    const int kh   = lane >> 4;            // K half-select (ISA 32-bit frag layout)
    const int n0   = gate * H_SIZE + (blockIdx.x << 4);   // z column base

    v8f acc[4] = {};                       // four 16x16 fp32 C tiles (M blocks)

    gemm_term<KQ0, 4 * KQ0>(A0 + kq * KQ0, W0 + (size_t)(kq * KQ0) * ZCOLS,
                            acc, lc, kh, n0);
    gemm_term<KQ1, 4 * KQ1>(A1 + kq * KQ1, W1 + (size_t)(kq * KQ1) * ZCOLS,
                            acc, lc, kh, n0);
    if constexpr (KQ2 > 0) {
        gemm_term<KQ2, 4 * KQ2>(A2 + kq * KQ2, W2 + (size_t)(kq * KQ2) * ZCOLS,
                                acc, lc, kh, n0);
    }

    // Spill z tiles to LDS: K-slice reduction + f/i/o/g recombine happen there.
    // C layout (ISA 7.12.2): VGPR r, lanes 0-15 -> M = 16*mb + r, N = lane;
    //                        lanes 16-31 -> M = 16*mb + 8 + r, N = lane-16.
    #pragma unroll
    for (int mb = 0; mb < 4; ++mb) {
        #pragma unroll
        for (int r = 0; r < 8; ++r) {
            const int M = mb * 16 + kh * 8 + r;
            zlds[kq][gate][M][lc] = acc[mb][r];
        }
    }
    __syncthreads();

    // Gate math + masked state update: 64 rows x 16 cols = 1024 elems, 2/thread.
    #pragma unroll
    for (int e = 0; e < 2; ++e) {
        const int idx  = e * 512 + tid;
        const int b    = idx >> 4;                      // batch row
        const int j    = idx & 15;                      // local col
        const int colg = (blockIdx.x << 4) + j;         // hidden index 0..1023

        float zf = bias[0 * H_SIZE + colg];
        float zi = bias[1 * H_SIZE + colg];
        float zo = bias[2 * H_SIZE + colg];
        float zg = bias[3 * H_SIZE + colg];
        #pragma unroll
        for (int q = 0; q < 4; ++q) {
            zf += zlds[q][0][b][j];
            zi += zlds[q][1][b][j];
            zo += zlds[q][2][b][j];
            zg += zlds[q][3][b][j];
        }

        const float cp = c_prev[b * H_SIZE + colg];
        const float hp = h_prev[b * H_SIZE + colg];

        const float c1 = sigmoidf_(zf) * cp + sigmoidf_(zi) * tanhf(zg);
        const float h1 = sigmoidf_(zo) * tanhf(c1);

        const bool m  = (t < length[b]);                // freeze when t >= length
        const float hn = m ? h1 : hp;
        const float cn = m ? c1 : cp;

        h_out[b * H_SIZE + colg] = hn;
        c_out[b * H_SIZE + colg] = cn;
        if (WRITE_OUT) out_t[b * H_SIZE + colg] = hn;
    }
}

// Copy initial h0/c0 (inputs, [3][B][H]) into parity-0 ping-pong buffers.
__global__ void init_states(const float* __restrict__ h0, const float* __restrict__ c0,
                            float* __restrict__ hws, float* __restrict__ cws)
{
    const int i = blockIdx.x * blockDim.x + threadIdx.x;
    if (i >= 3 * BH) return;
    const int l = i / BH;
    const int r = i % BH;
    hws[(l * 2 + 0) * BH + r] = h0[i];
    cws[(l * 2 + 0) * BH + r] = c0[i];
}

// Copy final Ht / C (at parity pfin) into the tail of d_out.
__global__ void copy_final(const float* __restrict__ hws, const float* __restrict__ cws,
                           float* __restrict__ out_ht, float* __restrict__ out_c, int pfin)
{
    const int i = blockIdx.x * blockDim.x + threadIdx.x;
    if (i >= 3 * BH) return;
    const int l = i / BH;
    const int r = i % BH;
    out_ht[i] = hws[(l * 2 + pfin) * BH + r];
    out_c[i]  = cws[(l * 2 + pfin) * BH + r];
}

extern "C" void kernel_launch(void* const* d_in, const int* in_sizes, int n_in,
                              void* d_out, int out_size, void* d_ws, size_t ws_size,
                              hipStream_t stream)
{
    const float* x      = (const float*)d_in[0];
    const int*   length = (const int*)  d_in[1];
    const float* h0     = (const float*)d_in[2];
    const float* c0     = (const float*)d_in[3];
    const float* Wbh0   = (const float*)d_in[4];
    const float* Whh0   = (const float*)d_in[5];
    const float* Wth0   = (const float*)d_in[6];
    const float* b0     = (const float*)d_in[7];
    const float* Wbh1   = (const float*)d_in[8];
    const float* Whh1   = (const float*)d_in[9];
    const float* Wth1   = (const float*)d_in[10];
    const float* b1     = (const float*)d_in[11];
    const float* Wih2   = (const float*)d_in[12];
    const float* Whh2   = (const float*)d_in[13];
    const float* b2     = (const float*)d_in[14];

    float* out = (float*)d_out;
    float* hws = (float*)d_ws;          // [3 layers][2 parity][B][H]
    float* cws = hws + 6 * BH;          // [3 layers][2 parity][B][H]  (total 3 MB)

    init_states<<<(3 * BH + 255) / 256, 256, 0, stream>>>(h0, c0, hws, cws);

    for (int t = 0; t < T_STEPS; ++t) {
        const int p = t & 1;            // read parity
        const int q = p ^ 1;            // write parity
        const float* x_t = x + (size_t)t * B_SIZE * I_SIZE;

        float* h0p = hws + (0 * 2 + p) * BH;  float* h0n = hws + (0 * 2 + q) * BH;
        float* h1p = hws + (1 * 2 + p) * BH;  float* h1n = hws + (1 * 2 + q) * BH;
        float* h2p = hws + (2 * 2 + p) * BH;  float* h2n = hws + (2 * 2 + q) * BH;
        float* c0p = cws + (0 * 2 + p) * BH;  float* c0n = cws + (0 * 2 + q) * BH;
        float* c1p = cws + (1 * 2 + p) * BH;  float* c1n = cws + (1 * 2 + q) * BH;
        float* c2p = cws + (2 * 2 + p) * BH;  float* c2n = cws + (2 * 2 + q) * BH;

        // layer 0: z = h0_prev@Whh0 + x_t@Wbh0 + h1_prev@Wth0 + b0
        lstm_layer_kernel<256, 128, 256, false><<<64, 512, 0, stream>>>(
            h0p, Whh0,  x_t, Wbh0,  h1p, Wth0,
            b0, h0p, c0p, h0n, c0n, nullptr, length, t);

        // layer 1: z = h1_prev@Whh1 + h0_new@Wbh1 + h2_prev@Wth1 + b1
        lstm_layer_kernel<256, 256, 256, false><<<64, 512, 0, stream>>>(
            h1p, Whh1,  h0n, Wbh1,  h2p, Wth1,
            b1, h1p, c1p, h1n, c1n, nullptr, length, t);

        // layer 2: z = h2_prev@Whh2 + h1_new@Wih2 + b2 ; writes output[t]
        lstm_layer_kernel<256, 256, 0, true><<<64, 512, 0, stream>>>(
            h2p, Whh2,  h1n, Wih2,  nullptr, nullptr,
            b2, h2p, c2p, h2n, c2n, out + (size_t)t * BH, length, t);
    }

    // Final Ht and C live at parity (T_STEPS & 1) == 0.
    copy_final<<<(3 * BH + 255) / 256, 256, 0, stream>>>(
        hws, cws,
        out + (size_t)T_STEPS * BH,
        out + (size_t)T_STEPS * BH + 3 * BH,
        T_STEPS & 1);
}